// SEASTAREmbedding_3539053052250
// MI455X (gfx1250) — compile-verified
//
#include <hip/hip_runtime.h>

typedef __attribute__((ext_vector_type(2))) float v2f;
typedef __attribute__((ext_vector_type(8))) float v8f;

#define BATCH     256
#define SEQ       512
#define DMODEL    256
#define TOK_TOTAL (BATCH * SEQ)
#define WAVES_PB  8
#define TOK_PW    16

// ln(10000)/256
#define PE_LN_SCALE 0.0359778921f
#define HALF_PI     1.57079632679f

__global__ __launch_bounds__(256) void seastar_embed_kernel(
    const float* __restrict__ src,        // [B,S,4]  (last col = category as float)
    const float* __restrict__ dist,       // [B,S,2]
    const int*   __restrict__ type_dist,  // [B,S,2]
    const float* __restrict__ env_dist,   // [B,S,4]
    const float* __restrict__ W_src,      // [3,64]
    const float* __restrict__ b_src,      // [3,64]
    const float* __restrict__ emb_src,    // [32,64]
    const float* __restrict__ W_dist,     // [2,64]
    const float* __restrict__ b_dist,     // [2,64]
    const float* __restrict__ emb_dist,   // [2,16,64]
    const float* __restrict__ W_env,      // [4,64]
    const float* __restrict__ b_env,      // [4,64]
    float* __restrict__ out)              // [3, B*S, 256] concatenated
{
    __shared__ float sWsrc[3 * 64];
    __shared__ float sBsrc[3 * 64];
    __shared__ float sEmbSrc[32 * 64];
    __shared__ float sWdist[2 * 64];
    __shared__ float sBdist[2 * 64];
    __shared__ float sEmbDist[2 * 16 * 64];
    __shared__ float sWenv[4 * 64];
    __shared__ float sBenv[4 * 64];
    __shared__ float sFreq[128];
    __shared__ float sSrc[WAVES_PB][16][4];
    __shared__ float sDist[WAVES_PB][16][2];
    __shared__ float sEnv[WAVES_PB][16][4];
    __shared__ int   sType[WAVES_PB][16][2];
    __shared__ int   sIdx[WAVES_PB][16];

    const int tid = threadIdx.x;

    // ---- cooperative parameter preload into LDS ----
    for (int i = tid; i < 3 * 64; i += 256) { sWsrc[i] = W_src[i]; sBsrc[i] = b_src[i]; }
    for (int i = tid; i < 32 * 64; i += 256) sEmbSrc[i] = emb_src[i];
    for (int i = tid; i < 2 * 64; i += 256) { sWdist[i] = W_dist[i]; sBdist[i] = b_dist[i]; }
    for (int i = tid; i < 2 * 16 * 64; i += 256) sEmbDist[i] = emb_dist[i];
    for (int i = tid; i < 4 * 64; i += 256) { sWenv[i] = W_env[i]; sBenv[i] = b_env[i]; }
    for (int i = tid; i < 128; i += 256)
        sFreq[i] = __expf((float)(-2 * i) * PE_LN_SCALE);

    const int wave = tid >> 5;
    const int lane = tid & 31;
    const int base = (blockIdx.x * WAVES_PB + wave) * TOK_PW; // first token of this wave

    // ---- per-wave token staging (lanes 0..15, one token each) ----
    if (lane < 16) {
        const int tok = base + lane;
        const float4 s4 = reinterpret_cast<const float4*>(src)[tok];
        sSrc[wave][lane][0] = s4.x; sSrc[wave][lane][1] = s4.y;
        sSrc[wave][lane][2] = s4.z; sSrc[wave][lane][3] = s4.w;
        sIdx[wave][lane] = (int)s4.w;
        const float2 d2 = reinterpret_cast<const float2*>(dist)[tok];
        sDist[wave][lane][0] = d2.x; sDist[wave][lane][1] = d2.y;
        const int2 t2 = reinterpret_cast<const int2*>(type_dist)[tok];
        sType[wave][lane][0] = t2.x; sType[wave][lane][1] = t2.y;
        const float4 e4 = reinterpret_cast<const float4*>(env_dist)[tok];
        sEnv[wave][lane][0] = e4.x; sEnv[wave][lane][1] = e4.y;
        sEnv[wave][lane][2] = e4.z; sEnv[wave][lane][3] = e4.w;
    }
    __syncthreads();

    // WMMA f32 16x16x4 lane mapping (wave32):
    //   A (16x4): lane L holds row m = L&15; VGPR j = A[m][2*(L>>4)+j]
    //   B (4x16): lane L holds col n = L&15; VGPR j = B[2*(L>>4)+j][n]
    //   C/D     : lane L holds col n = L&15; VGPR j = C[8*(L>>4)+j][n]
    const int n  = lane & 15;
    const int h  = lane >> 4;
    const int m0 = 8 * h;                      // first C/D row owned by this lane
    const int r0 = 2 * h, r1 = 2 * h + 1;      // B rows owned by this lane
    const int s0 = (base & (SEQ - 1)) + m0;    // seq pos of row M=m0 (S % 16 == 0)
    const float phase = (n & 1) ? HALF_PI : 0.0f;  // cos(x) == sin(x + pi/2)

    // A operands are tile-invariant
    const v2f A_src = { sSrc[wave][n][2 * h], sSrc[wave][n][2 * h + 1] };
    const v2f A_dst = { (h == 0) ? sDist[wave][n][0] : 0.0f,
                        (h == 0) ? sDist[wave][n][1] : 0.0f };
    const v2f A_env = { sEnv[wave][n][2 * h], sEnv[wave][n][2 * h + 1] };

    const size_t BSD = (size_t)TOK_TOTAL * DMODEL;
    const size_t rowOff = (size_t)(base + m0) * DMODEL;
    float* __restrict__ outS = out + rowOff;
    float* __restrict__ outD = out + BSD + rowOff;
    float* __restrict__ outE = out + 2 * BSD + rowOff;

#pragma unroll 2
    for (int t = 0; t < 16; ++t) {
        const int f = t >> 2;          // feature block: WAVE-UNIFORM per tile
        const int d = 16 * t + n;      // output dim handled by this lane's column
        const int c = 16 * (t & 3) + n;  // == d & 63
        const float freq = sFreq[d >> 1];

        // positional encoding for this lane's 8 (token, dim) cells
        float pe[8];
#pragma unroll
        for (int j = 0; j < 8; ++j)
            pe[j] = __sinf(fmaf((float)(s0 + j), freq, phase));

        // ---------------- SRC: linear (f<3) + category embedding (f==3) ------
        if (f < 3) {               // uniform branch
            const float wv = sWsrc[f * 64 + c];
            const v2f Bv = { (r0 == f) ? wv : 0.0f,
                             (r1 == f) ? wv : 0.0f };
            const float bias = sBsrc[f * 64 + c];
            v8f C;
#pragma unroll
            for (int j = 0; j < 8; ++j) C[j] = pe[j] + bias;
            const v8f Dv = __builtin_amdgcn_wmma_f32_16x16x4_f32(
                false, A_src, false, Bv, (short)0, C, false, false);
            float* __restrict__ p = outS + d;
#pragma unroll
            for (int j = 0; j < 8; ++j) p[(size_t)j * DMODEL] = Dv[j];
        } else {                   // pure embedding block: B==0 -> D == C, skip WMMA
            float* __restrict__ p = outS + d;
#pragma unroll
            for (int j = 0; j < 8; ++j)
                p[(size_t)j * DMODEL] = pe[j] + sEmbSrc[sIdx[wave][m0 + j] * 64 + c];
        }

        // ---------------- DIST: [emb_f(type), lin_f(dist)] interleaved -------
        {
            const int fe = f >> 1;     // feature 0/1 (uniform)
            if (f & 1) {               // odd 64-block = linear (uniform branch)
                const float wv = sWdist[fe * 64 + c];
                const v2f Bv = { (r0 == fe) ? wv : 0.0f,
                                 (r1 == fe) ? wv : 0.0f };
                const float bias = sBdist[fe * 64 + c];
                v8f C;
#pragma unroll
                for (int j = 0; j < 8; ++j) C[j] = pe[j] + bias;
                const v8f Dv = __builtin_amdgcn_wmma_f32_16x16x4_f32(
                    false, A_dst, false, Bv, (short)0, C, false, false);
                float* __restrict__ p = outD + d;
#pragma unroll
                for (int j = 0; j < 8; ++j) p[(size_t)j * DMODEL] = Dv[j];
            } else {                   // even 64-block = type embedding, D == C
                float* __restrict__ p = outD + d;
#pragma unroll
                for (int j = 0; j < 8; ++j)
                    p[(size_t)j * DMODEL] =
                        pe[j] + sEmbDist[(fe * 16 + sType[wave][m0 + j][fe]) * 64 + c];
            }
        }

        // ---------------- ENV: pure per-feature linear ------------------------
        {
            const float wv = sWenv[f * 64 + c];
            const v2f Bv = { (r0 == f) ? wv : 0.0f,
                             (r1 == f) ? wv : 0.0f };
            const float bias = sBenv[f * 64 + c];
            v8f C;
#pragma unroll
            for (int j = 0; j < 8; ++j) C[j] = pe[j] + bias;
            const v8f Dv = __builtin_amdgcn_wmma_f32_16x16x4_f32(
                false, A_env, false, Bv, (short)0, C, false, false);
            float* __restrict__ p = outE + d;
#pragma unroll
            for (int j = 0; j < 8; ++j) p[(size_t)j * DMODEL] = Dv[j];
        }
    }
}

extern "C" void kernel_launch(void* const* d_in, const int* in_sizes, int n_in,
                              void* d_out, int out_size, void* d_ws, size_t ws_size,
                              hipStream_t stream) {
    (void)in_sizes; (void)n_in; (void)out_size; (void)d_ws; (void)ws_size;
    const float* src       = (const float*)d_in[0];
    const float* dist      = (const float*)d_in[1];
    const int*   type_dist = (const int*)  d_in[2];
    const float* env_dist  = (const float*)d_in[3];
    const float* W_src     = (const float*)d_in[4];
    const float* b_src     = (const float*)d_in[5];
    const float* emb_src   = (const float*)d_in[6];
    const float* W_dist    = (const float*)d_in[7];
    const float* b_dist    = (const float*)d_in[8];
    const float* emb_dist  = (const float*)d_in[9];
    const float* W_env     = (const float*)d_in[10];
    const float* b_env     = (const float*)d_in[11];
    float* out = (float*)d_out;

    const int groups = TOK_TOTAL / TOK_PW;          // 8192 waves of work
    const int blocks = groups / WAVES_PB;           // 1024 blocks x 256 threads
    seastar_embed_kernel<<<blocks, 256, 0, stream>>>(
        src, dist, type_dist, env_dist,
        W_src, b_src, emb_src,
        W_dist, b_dist, emb_dist,
        W_env, b_env, out);
}